// GNNRecommender_1726576853567
// MI455X (gfx1250) — compile-verified
//
#include <hip/hip_runtime.h>
#include <math.h>

#define NN 100000
#define EE 1600000
#define HH 64
#define BN_EPS 1e-5f
#define L2REG 1e-5f

typedef float v2f __attribute__((ext_vector_type(2)));
typedef float v8f __attribute__((ext_vector_type(8)));

static __host__ int cdiv(int a, int b) { return (a + b - 1) / b; }

// ---------------- utility ----------------
__global__ void k_fill(float* p, int n, float v) {
    int i = blockIdx.x * blockDim.x + threadIdx.x;
    if (i < n) p[i] = v;
}

// deg += 1 per incoming edge (deg pre-filled with 1.0 for self loop)
__global__ void k_deg(const int* __restrict__ dst, float* __restrict__ deg) {
    int e = blockIdx.x * blockDim.x + threadIdx.x;
    if (e < EE) unsafeAtomicAdd(&deg[dst[e]], 1.0f);
}

__global__ void k_dinv(float* d) {
    int i = blockIdx.x * blockDim.x + threadIdx.x;
    if (i < NN) { float v = d[i]; d[i] = (v > 0.f) ? rsqrtf(v) : 0.f; }
}

// h0 = emb[x]
__global__ void k_gather(const int* __restrict__ x, const float* __restrict__ emb,
                         float* __restrict__ h) {
    int i = blockIdx.x * blockDim.x + threadIdx.x;
    if (i < NN * HH) {
        int node = i >> 6, c = i & 63;
        h[i] = emb[(size_t)x[node] * HH + c];
    }
}

// ---------------- fp32 WMMA GEMM: O[N,64] = A[N,64] @ W[64,64] ----------------
// 4 waves/block, each wave owns 16 rows x 64 cols. K=64 via 16 steps of
// v_wmma_f32_16x16x4_f32 across 4 column tiles. W staged in LDS.
__global__ void __launch_bounds__(128) k_gemm64(const float* __restrict__ A,
                                                const float* __restrict__ W,
                                                float* __restrict__ O) {
    __shared__ float Wl[64 * 64];
    int tid = threadIdx.x;
    for (int i = tid; i < 64 * 64; i += 128) Wl[i] = W[i];
    __syncthreads();

    int wave = tid >> 5, lane = tid & 31;
    int rowBase = blockIdx.x * 64 + wave * 16;
    int m = lane & 15;       // M (for A load) / N (for B load & C store)
    int kp = lane >> 4;      // K-pair selector (0 -> K{0,1}, 1 -> K{2,3})

    int arow = rowBase + m;
    if (arow > NN - 1) arow = NN - 1;                 // clamp loads, guard stores
    const float* Ap = A + (size_t)arow * 64;

    v8f acc0 = {}, acc1 = {}, acc2 = {}, acc3 = {};

#pragma unroll
    for (int k = 0; k < 64; k += 4) {
        int ka = k + kp * 2;                          // even -> 8B-aligned
        v2f a = *(const v2f*)(Ap + ka);               // A[m, ka], A[m, ka+1]
        const float* w0 = &Wl[ka * 64 + m];
        const float* w1 = &Wl[(ka + 1) * 64 + m];
        v2f b0, b1, b2, b3;
        b0.x = w0[0];  b0.y = w1[0];
        b1.x = w0[16]; b1.y = w1[16];
        b2.x = w0[32]; b2.y = w1[32];
        b3.x = w0[48]; b3.y = w1[48];
        acc0 = __builtin_amdgcn_wmma_f32_16x16x4_f32(false, a, false, b0, (short)0, acc0, false, false);
        acc1 = __builtin_amdgcn_wmma_f32_16x16x4_f32(false, a, false, b1, (short)0, acc1, false, false);
        acc2 = __builtin_amdgcn_wmma_f32_16x16x4_f32(false, a, false, b2, (short)0, acc2, false, false);
        acc3 = __builtin_amdgcn_wmma_f32_16x16x4_f32(false, a, false, b3, (short)0, acc3, false, false);
    }

    // C/D layout: VGPR v -> M = v + 8*kp, N = lane&15
    int mB = rowBase + kp * 8;
#pragma unroll
    for (int v = 0; v < 8; ++v) {
        int r = mB + v;
        if (r < NN) {
            size_t ro = (size_t)r * 64 + m;
            O[ro]      = acc0[v];
            O[ro + 16] = acc1[v];
            O[ro + 32] = acc2[v];
            O[ro + 48] = acc3[v];
        }
    }
}

// agg = hw * dinv[node]^2   (self-loop contribution; also initializes agg)
__global__ void k_selfloop(const float* __restrict__ hw, const float* __restrict__ dinv,
                           float* __restrict__ agg) {
    int i = blockIdx.x * blockDim.x + threadIdx.x;
    if (i < NN * HH) {
        float di = dinv[i >> 6];
        agg[i] = hw[i] * di * di;
    }
}

// agg[dst] += hw[src] * dinv[src]*dinv[dst]  (64 lanes per edge, native f32 atomics)
__global__ void k_edge_scatter(const int* __restrict__ src, const int* __restrict__ dst,
                               const float* __restrict__ hw, const float* __restrict__ dinv,
                               float* __restrict__ agg) {
    int t = blockIdx.x * blockDim.x + threadIdx.x;
    int e = t >> 6, j = t & 63;
    if (e < EE) {
        int s = src[e], d = dst[e];
        float nrm = dinv[s] * dinv[d];
        unsafeAtomicAdd(&agg[(size_t)d * 64 + j], hw[(size_t)s * 64 + j] * nrm);
    }
}

// per-channel sum / sumsq -> stats[0..63], stats[64..127]
__global__ void k_bn_stats(const float* __restrict__ a, float* __restrict__ stats) {
    int c = threadIdx.x & 63;
    int rowStart = blockIdx.x * (blockDim.x >> 6) + (threadIdx.x >> 6);
    int rowStride = gridDim.x * (blockDim.x >> 6);
    float s = 0.f, q = 0.f;
    for (int r = rowStart; r < NN; r += rowStride) {
        float v = a[(size_t)r * 64 + c];
        s += v; q += v * v;
    }
    unsafeAtomicAdd(&stats[c], s);
    unsafeAtomicAdd(&stats[64 + c], q);
}

// scale/shift -> stats[128..191], stats[192..255]  (conv bias cancels inside BN)
__global__ void k_bn_finalize(float* stats, const float* __restrict__ g,
                              const float* __restrict__ be) {
    int c = threadIdx.x;
    if (c < 64) {
        float mean = stats[c] * (1.0f / NN);
        float var = stats[64 + c] * (1.0f / NN) - mean * mean;
        float sc = g[c] * rsqrtf(var + BN_EPS);
        stats[128 + c] = sc;
        stats[192 + c] = be[c] - mean * sc;
    }
}

__global__ void k_bn_apply_relu(const float* __restrict__ a, const float* __restrict__ stats,
                                float* __restrict__ o) {
    int i = blockIdx.x * blockDim.x + threadIdx.x;
    if (i < NN * HH) {
        int c = i & 63;
        float y = a[i] * stats[128 + c] + stats[192 + c];
        o[i] = fmaxf(y, 0.f);
    }
}

// h3 = relu(agg + b3 + h_res)
__global__ void k_res_relu(const float* __restrict__ a, const float* __restrict__ b3,
                           const float* __restrict__ res, float* __restrict__ o) {
    int i = blockIdx.x * blockDim.x + threadIdx.x;
    if (i < NN * HH) {
        float y = a[i] + b3[i & 63] + res[i];
        o[i] = fmaxf(y, 0.f);
    }
}

// ratings[e] = dot(h[src], Wout[0:64]) + dot(h[dst], Wout[64:128]) + bout
__global__ void k_decode(const int* __restrict__ src, const int* __restrict__ dst,
                         const float* __restrict__ h, const float* __restrict__ Wout,
                         const float* __restrict__ bout, float* __restrict__ out) {
    int t = blockIdx.x * blockDim.x + threadIdx.x;
    int e = t >> 5, lane = t & 31;
    if (e >= EE) return;
    int s = src[e], d = dst[e];
    v2f hs = *(const v2f*)(h + (size_t)s * 64 + lane * 2);
    v2f hd = *(const v2f*)(h + (size_t)d * 64 + lane * 2);
    v2f ws = *(const v2f*)(Wout + lane * 2);
    v2f wd = *(const v2f*)(Wout + 64 + lane * 2);
    float p = hs.x * ws.x + hs.y * ws.y + hd.x * wd.x + hd.y * wd.y;
#pragma unroll
    for (int o = 16; o > 0; o >>= 1) p += __shfl_xor(p, o, 32);
    if (lane == 0) out[e] = p + bout[0];
}

// per-tensor sum of squares into ssq[slot]
__global__ void k_sumsq(const float* __restrict__ p, int n, float* __restrict__ slot) {
    __shared__ float red[256];
    float s = 0.f;
    for (int i = blockIdx.x * blockDim.x + threadIdx.x; i < n; i += gridDim.x * blockDim.x) {
        float v = p[i]; s += v * v;
    }
    red[threadIdx.x] = s;
    __syncthreads();
    for (int o = 128; o > 0; o >>= 1) {
        if (threadIdx.x < o) red[threadIdx.x] += red[threadIdx.x + o];
        __syncthreads();
    }
    if (threadIdx.x == 0) unsafeAtomicAdd(slot, red[0]);
}

__global__ void k_l2_final(const float* __restrict__ ssq, float* __restrict__ out) {
    if (threadIdx.x == 0) {
        float s = 0.f;
        for (int i = 0; i < 13; ++i) s += sqrtf(ssq[i]);
        *out = s * L2REG;
    }
}

extern "C" void kernel_launch(void* const* d_in, const int* in_sizes, int n_in,
                              void* d_out, int out_size, void* d_ws, size_t ws_size,
                              hipStream_t stream) {
    const int*   x    = (const int*)d_in[0];
    const int*   ei   = (const int*)d_in[1];
    const float* emb  = (const float*)d_in[2];
    const float* W1   = (const float*)d_in[3];
    // b1/b2 (d_in[4], d_in[6]) cancel inside BN; included only in the L2 term
    const float* W2   = (const float*)d_in[5];
    const float* W3   = (const float*)d_in[7];
    const float* b3   = (const float*)d_in[8];
    const float* g1   = (const float*)d_in[9];
    const float* be1  = (const float*)d_in[10];
    const float* g2   = (const float*)d_in[11];
    const float* be2  = (const float*)d_in[12];
    const float* Wout = (const float*)d_in[13];
    const float* bout = (const float*)d_in[14];

    const int* src = ei;
    const int* dst = ei + EE;

    float* ws    = (float*)d_ws;
    float* dinv  = ws;                       // N
    float* stats = ws + NN;                  // 256
    float* ssq   = ws + NN + 256;            // 16 (13 used)
    float* bufH  = ws + NN + 320;            // N*64  (h input / h2 / h3)
    float* bufW  = bufH + (size_t)NN * HH;   // N*64  (hw)
    float* bufA  = bufW + (size_t)NN * HH;   // N*64  (aggregate)
    float* bufR  = bufA + (size_t)NN * HH;   // N*64  (h1 residual)
    // total ~103 MB of workspace

    float* ratings = (float*)d_out;          // [E], then l2 scalar at [E]

    const int TB = 256;
    int gN   = cdiv(NN, TB);
    int gNH  = cdiv(NN * HH, TB);
    int gE   = cdiv(EE, TB);
    int gE64 = cdiv(EE * 64, TB);
    int gE32 = cdiv(EE * 32, TB);
    int gGemm = cdiv(NN, 64);

    // degree -> dinv
    k_fill<<<gN, TB, 0, stream>>>(dinv, NN, 1.0f);   // self-loop
    k_deg<<<gE, TB, 0, stream>>>(dst, dinv);
    k_dinv<<<gN, TB, 0, stream>>>(dinv);

    // h0 = emb[x]
    k_gather<<<gNH, TB, 0, stream>>>(x, emb, bufH);

    // ---- layer 1: h1 = relu(BN(gcn(h0,W1))) -> bufR (residual) ----
    k_gemm64<<<gGemm, 128, 0, stream>>>(bufH, W1, bufW);
    k_selfloop<<<gNH, TB, 0, stream>>>(bufW, dinv, bufA);
    k_edge_scatter<<<gE64, TB, 0, stream>>>(src, dst, bufW, dinv, bufA);
    k_fill<<<1, 128, 0, stream>>>(stats, 128, 0.0f);
    k_bn_stats<<<512, TB, 0, stream>>>(bufA, stats);
    k_bn_finalize<<<1, 64, 0, stream>>>(stats, g1, be1);
    k_bn_apply_relu<<<gNH, TB, 0, stream>>>(bufA, stats, bufR);

    // ---- layer 2: h2 = relu(BN(gcn(h1,W2))) -> bufH ----
    k_gemm64<<<gGemm, 128, 0, stream>>>(bufR, W2, bufW);
    k_selfloop<<<gNH, TB, 0, stream>>>(bufW, dinv, bufA);
    k_edge_scatter<<<gE64, TB, 0, stream>>>(src, dst, bufW, dinv, bufA);
    k_fill<<<1, 128, 0, stream>>>(stats, 128, 0.0f);
    k_bn_stats<<<512, TB, 0, stream>>>(bufA, stats);
    k_bn_finalize<<<1, 64, 0, stream>>>(stats, g2, be2);
    k_bn_apply_relu<<<gNH, TB, 0, stream>>>(bufA, stats, bufH);

    // ---- layer 3: h3 = relu(gcn(h2,W3) + b3 + h1) -> bufH ----
    k_gemm64<<<gGemm, 128, 0, stream>>>(bufH, W3, bufW);
    k_selfloop<<<gNH, TB, 0, stream>>>(bufW, dinv, bufA);
    k_edge_scatter<<<gE64, TB, 0, stream>>>(src, dst, bufW, dinv, bufA);
    k_res_relu<<<gNH, TB, 0, stream>>>(bufA, b3, bufR, bufH);

    // ---- edge decode ----
    k_decode<<<gE32, TB, 0, stream>>>(src, dst, bufH, Wout, bout, ratings);

    // ---- L2 regularization term: sum of 13 per-tensor norms ----
    k_fill<<<1, 16, 0, stream>>>(ssq, 16, 0.0f);
    for (int i = 0; i < 13; ++i) {
        const float* p = (const float*)d_in[2 + i];
        int n = in_sizes[2 + i];
        int g = cdiv(n, TB); if (g > 1024) g = 1024;
        k_sumsq<<<g, TB, 0, stream>>>(p, n, &ssq[i]);
    }
    k_l2_final<<<1, 32, 0, stream>>>(ssq, ratings + EE);
}